// AdaptiveECE_33303176413863
// MI455X (gfx1250) — compile-verified
//
#include <hip/hip_runtime.h>
#include <stdint.h>

#define NROWS 1000000
#define NCOLS 128
#define NB    15
#define NTGT  30

// ---- workspace byte offsets ----
#define OFF_CONF   0ull                 // NROWS f32
#define OFF_ACC    4000000ull           // NROWS u8
#define OFF_H1     5000064ull           // 2048 u32
#define OFF_H2     5008256ull           // NTGT*2048 u32
#define OFF_H3     5254016ull           // NTGT*1024 u32
#define OFF_BINS   5376896ull           // cnt u32[16] | asum u32[16] | csum u64[16]
#define ZERO_END   5377152ull
#define OFF_PFX    5377152ull           // NTGT u32
#define OFF_RR     5377280ull           // NTGT u32
#define OFF_EDGES  5377408ull           // 16 f32
#define ZERO_WORDS ((unsigned)((ZERO_END - OFF_H1) / 4ull))

// K0: zero all histograms/accumulators, set target ranks (fp32 linspace arithmetic)
__global__ __launch_bounds__(256) void ece_init(unsigned* zb, unsigned* pfx, unsigned* rr)
{
    unsigned i = blockIdx.x * 256u + threadIdx.x;
    unsigned stride = gridDim.x * 256u;
    for (unsigned j = i; j < ZERO_WORDS; j += stride) zb[j] = 0u;
    if (blockIdx.x == 0 && threadIdx.x < NTGT) {
        int t = (int)threadIdx.x;
        unsigned rank;
        if (t == 0)           rank = 0u;
        else if (t == NTGT-1) rank = NROWS - 1u;
        else {
            int k = (t + 1) >> 1;                       // t=2k-1 -> i0, t=2k -> i0+1
            float qk = (float)k * (1000000.0f / 15.0f); // fp32, matches linspace step
            unsigned i0 = (unsigned)floorf(qk);
            rank = i0 + ((t & 1) ? 0u : 1u);
        }
        pfx[t] = 0u;
        rr[t]  = rank;
    }
}

// per-row reduction: max/argmax/sum-exp via wave32 butterflies
__device__ __forceinline__ void process_row(const float* buf, unsigned row, int lane,
                                            const int* __restrict__ labels,
                                            float* __restrict__ conf,
                                            unsigned char* __restrict__ acc)
{
    float4 v = *(const float4*)(buf + lane * 4);
    float m = v.x; int mi = 0;
    if (v.y > m) { m = v.y; mi = 1; }
    if (v.z > m) { m = v.z; mi = 2; }
    if (v.w > m) { m = v.w; mi = 3; }
    int gidx = lane * 4 + mi;
    #pragma unroll
    for (int off = 16; off > 0; off >>= 1) {
        float om = __shfl_xor(m, off, 32);
        int   oi = __shfl_xor(gidx, off, 32);
        if (om > m || (om == m && oi < gidx)) { m = om; gidx = oi; }
    }
    float s = __expf(v.x - m) + __expf(v.y - m) + __expf(v.z - m) + __expf(v.w - m);
    #pragma unroll
    for (int off = 16; off > 0; off >>= 1) s += __shfl_xor(s, off, 32);
    if (lane == 0) {
        conf[row] = 1.0f / s;
        acc[row]  = (gidx == labels[row]) ? (unsigned char)1 : (unsigned char)0;
    }
}

// K1: one wave handles 8 consecutive rows. 8-deep async pipeline:
// issue 8 global_load_async_to_lds_b128 (offset: is added to BOTH the LDS and
// global addresses per the ISA), then drain with staged s_wait_asynccnt 7..0.
// 4 KB in flight per wave keeps HBM at the roofline; th:TH_LOAD_NT since the
// 512 MB logits stream is touched exactly once.
__global__ __launch_bounds__(256) void ece_rows(const float* __restrict__ logits,
                                                const int* __restrict__ labels,
                                                float* __restrict__ conf,
                                                unsigned char* __restrict__ acc)
{
    __shared__ __align__(16) float smem[8 * 8 * NCOLS];   // 8 waves x 8 rows x 512B = 32 KB
    const int lane = threadIdx.x & 31;
    const int wv   = threadIdx.x >> 5;
    const unsigned base_row = (blockIdx.x * 8u + (unsigned)wv) * 8u;
    const float* rowp = logits + (size_t)base_row * NCOLS;

    unsigned ldsOff = (unsigned)(uintptr_t)&smem[wv * 1024 + lane * 4]; // flat low 32b == DS addr
    unsigned long long g = (unsigned long long)(uintptr_t)rowp;        // wave-uniform
    unsigned glo = __builtin_amdgcn_readfirstlane((unsigned)(g & 0xffffffffull));
    unsigned ghi = __builtin_amdgcn_readfirstlane((unsigned)(g >> 32));
    unsigned long long sbase = ((unsigned long long)ghi << 32) | (unsigned long long)glo;
    unsigned voff = (unsigned)(lane << 4);                             // 16B per lane

#define ECE_ISSUE(OFFB) \
    asm volatile("global_load_async_to_lds_b128 %0, %1, %2 offset:" #OFFB " th:TH_LOAD_NT" \
                 :: "v"(ldsOff), "v"(voff), "s"(sbase) : "memory")
    ECE_ISSUE(0);    ECE_ISSUE(512);  ECE_ISSUE(1024); ECE_ISSUE(1536);
    ECE_ISSUE(2048); ECE_ISSUE(2560); ECE_ISSUE(3072); ECE_ISSUE(3584);
#undef ECE_ISSUE

#define ECE_STEP(R, LEFT) \
    asm volatile("s_wait_asynccnt " #LEFT ::: "memory"); \
    process_row(&smem[wv * 1024 + (R) * NCOLS], base_row + (R), lane, labels, conf, acc)
    ECE_STEP(0, 7); ECE_STEP(1, 6); ECE_STEP(2, 5); ECE_STEP(3, 4);
    ECE_STEP(4, 3); ECE_STEP(5, 2); ECE_STEP(6, 1); ECE_STEP(7, 0);
#undef ECE_STEP
}

// K2: level-1 histogram (top 11 bits of float bit pattern)
__global__ __launch_bounds__(256) void ece_hist1(const float* __restrict__ conf, unsigned* __restrict__ H1)
{
    __shared__ unsigned h[2048];
    for (int i = threadIdx.x; i < 2048; i += 256) h[i] = 0u;
    __syncthreads();
    for (int i = blockIdx.x * 256 + threadIdx.x; i < NROWS; i += gridDim.x * 256) {
        unsigned key = __float_as_uint(conf[i]);
        atomicAdd(&h[key >> 21], 1u);
    }
    __syncthreads();
    for (int b = threadIdx.x; b < 2048; b += 256) if (h[b]) atomicAdd(&H1[b], h[b]);
}

// K3: locate all 30 ascending ranks in the 2048-bin histogram
__global__ __launch_bounds__(256) void ece_scan1(const unsigned* __restrict__ H1,
                                                 unsigned* __restrict__ pfx, unsigned* __restrict__ rr)
{
    __shared__ unsigned h[2048];
    __shared__ unsigned ps[256];
    for (int i = threadIdx.x; i < 2048; i += 256) h[i] = H1[i];
    __syncthreads();
    unsigned s = 0;
    #pragma unroll
    for (int j = 0; j < 8; j++) s += h[threadIdx.x * 8 + j];
    ps[threadIdx.x] = s;
    __syncthreads();
    if (threadIdx.x == 0) {
        unsigned r[NTGT];
        for (int t = 0; t < NTGT; t++) r[t] = rr[t];
        unsigned cum = 0; int tp = 0;
        for (int ch = 0; ch < 256 && tp < NTGT; ch++) {
            unsigned c = ps[ch];
            while (tp < NTGT && r[tp] < cum + c) {
                unsigned cc = cum;
                for (int b = ch * 8; ; b++) {
                    unsigned cb = h[b];
                    if (r[tp] < cc + cb) { pfx[tp] = (unsigned)b << 21; rr[tp] = r[tp] - cc; break; }
                    cc += cb;
                }
                tp++;
            }
            cum += c;
        }
    }
}

// K4: level-2 histogram (middle 11 bits) for elements matching any target's top bin
__global__ __launch_bounds__(256) void ece_hist2(const float* __restrict__ conf,
                                                 const unsigned* __restrict__ pfx, unsigned* __restrict__ H2)
{
    __shared__ unsigned p[NTGT];
    if (threadIdx.x < NTGT) p[threadIdx.x] = pfx[threadIdx.x];
    __syncthreads();
    for (int i = blockIdx.x * 256 + threadIdx.x; i < NROWS; i += gridDim.x * 256) {
        unsigned key = __float_as_uint(conf[i]);
        unsigned top = key & 0xFFE00000u;
        unsigned mid = (key >> 10) & 0x7FFu;
        #pragma unroll
        for (int t = 0; t < NTGT; t++)
            if (p[t] == top) atomicAdd(&H2[t * 2048 + mid], 1u);
    }
}

// K5: per-target locate rank within its 2048-bin mid histogram
__global__ __launch_bounds__(256) void ece_scan2(const unsigned* __restrict__ H2,
                                                 unsigned* __restrict__ pfx, unsigned* __restrict__ rr)
{
    int t = blockIdx.x;
    __shared__ unsigned h[2048];
    __shared__ unsigned ps[256];
    for (int i = threadIdx.x; i < 2048; i += 256) h[i] = H2[t * 2048 + i];
    __syncthreads();
    unsigned s = 0;
    #pragma unroll
    for (int j = 0; j < 8; j++) s += h[threadIdx.x * 8 + j];
    ps[threadIdx.x] = s;
    __syncthreads();
    if (threadIdx.x == 0) {
        unsigned r = rr[t], cum = 0; int ch = 0;
        for (; ch < 256; ch++) { unsigned c = ps[ch]; if (r < cum + c) break; cum += c; }
        for (int b = ch * 8; ; b++) {
            unsigned cb = h[b];
            if (r < cum + cb) { pfx[t] |= (unsigned)b << 10; rr[t] = r - cum; break; }
            cum += cb;
        }
    }
}

// K6: level-3 histogram (low 10 bits) for elements matching target's top-22 bits
__global__ __launch_bounds__(256) void ece_hist3(const float* __restrict__ conf,
                                                 const unsigned* __restrict__ pfx, unsigned* __restrict__ H3)
{
    __shared__ unsigned p[NTGT];
    if (threadIdx.x < NTGT) p[threadIdx.x] = pfx[threadIdx.x];
    __syncthreads();
    for (int i = blockIdx.x * 256 + threadIdx.x; i < NROWS; i += gridDim.x * 256) {
        unsigned key = __float_as_uint(conf[i]);
        unsigned top = key & 0xFFFFFC00u;
        unsigned lo  = key & 0x3FFu;
        #pragma unroll
        for (int t = 0; t < NTGT; t++)
            if (p[t] == top) atomicAdd(&H3[t * 1024 + lo], 1u);
    }
}

// K7: final 10 bits -> pfx[t] becomes the exact bit pattern of sorted_conf[rank_t]
__global__ __launch_bounds__(256) void ece_scan3(const unsigned* __restrict__ H3,
                                                 unsigned* __restrict__ pfx, unsigned* __restrict__ rr)
{
    int t = blockIdx.x;
    __shared__ unsigned h[1024];
    __shared__ unsigned ps[256];
    for (int i = threadIdx.x; i < 1024; i += 256) h[i] = H3[t * 1024 + i];
    __syncthreads();
    unsigned s = 0;
    #pragma unroll
    for (int j = 0; j < 4; j++) s += h[threadIdx.x * 4 + j];
    ps[threadIdx.x] = s;
    __syncthreads();
    if (threadIdx.x == 0) {
        unsigned r = rr[t], cum = 0; int ch = 0;
        for (; ch < 256; ch++) { unsigned c = ps[ch]; if (r < cum + c) break; cum += c; }
        for (int b = ch * 4; ; b++) {
            unsigned cb = h[b];
            if (r < cum + cb) { pfx[t] |= (unsigned)b; break; }
            cum += cb;
        }
    }
}

// K8: edges[0..15] via jnp.interp semantics (fp32 q = k * (1e6/15))
__global__ void ece_edges(const unsigned* __restrict__ keys, float* __restrict__ edges)
{
    if (threadIdx.x == 0) {
        edges[0]  = __uint_as_float(keys[0]);        // rank 0  (min)
        edges[15] = __uint_as_float(keys[NTGT - 1]); // rank N-1 (max, q clamps)
        for (int k = 1; k <= 14; k++) {
            float qk   = (float)k * (1000000.0f / 15.0f);
            float fi   = floorf(qk);
            float frac = qk - fi;
            float lo = __uint_as_float(keys[2 * k - 1]);
            float hi = __uint_as_float(keys[2 * k]);
            edges[k] = lo + (hi - lo) * frac;
        }
    }
}

// K9: bin every element, accumulate cnt / acc-sum / fixed-point conf-sum (deterministic)
__global__ __launch_bounds__(256) void ece_bin(const float* __restrict__ conf,
                                               const unsigned char* __restrict__ acc,
                                               const float* __restrict__ edges,
                                               unsigned* __restrict__ cnt,
                                               unsigned* __restrict__ asum,
                                               unsigned long long* __restrict__ csum)
{
    __shared__ float e[16];
    __shared__ unsigned lc[16], la[16];
    __shared__ unsigned long long ls[16];
    if (threadIdx.x < 16) {
        e[threadIdx.x]  = edges[threadIdx.x];
        lc[threadIdx.x] = 0u; la[threadIdx.x] = 0u; ls[threadIdx.x] = 0ull;
    }
    __syncthreads();
    for (int i = blockIdx.x * 256 + threadIdx.x; i < NROWS; i += gridDim.x * 256) {
        float c = conf[i];
        if (c > e[0] && c <= e[15]) {              // valid: (edges[0], edges[15]]
            int bin = 0;
            #pragma unroll
            for (int j = 1; j <= 14; j++) bin += (c > e[j]) ? 1 : 0;  // searchsorted left
            atomicAdd(&lc[bin], 1u);
            atomicAdd(&la[bin], (unsigned)acc[i]);
            atomicAdd(&ls[bin], (unsigned long long)(c * 1073741824.0f + 0.5f)); // conf * 2^30
        }
    }
    __syncthreads();
    if (threadIdx.x < 15 && lc[threadIdx.x]) {
        atomicAdd(&cnt[threadIdx.x],  lc[threadIdx.x]);
        atomicAdd(&asum[threadIdx.x], la[threadIdx.x]);
        atomicAdd(&csum[threadIdx.x], ls[threadIdx.x]);
    }
}

// K10: final ECE
__global__ void ece_final(const unsigned* __restrict__ cnt,
                          const unsigned* __restrict__ asum,
                          const unsigned long long* __restrict__ csum,
                          float* __restrict__ out)
{
    if (threadIdx.x == 0 && blockIdx.x == 0) {
        double ece = 0.0;
        for (int b = 0; b < NB; b++) {
            unsigned c = cnt[b];
            if (c > 0) {
                double cs = (double)csum[b] * (1.0 / 1073741824.0);
                double ca = cs / (double)c;
                double aa = (double)asum[b] / (double)c;
                ece += fabs(ca - aa) * ((double)c / 1000000.0);
            }
        }
        out[0] = (float)ece;
    }
}

extern "C" void kernel_launch(void* const* d_in, const int* in_sizes, int n_in,
                              void* d_out, int out_size, void* d_ws, size_t ws_size,
                              hipStream_t stream)
{
    (void)in_sizes; (void)n_in; (void)out_size; (void)ws_size;
    const float* logits = (const float*)d_in[0];
    const int*   labels = (const int*)d_in[1];
    char* ws = (char*)d_ws;

    float*              conf  = (float*)(ws + OFF_CONF);
    unsigned char*      acc   = (unsigned char*)(ws + OFF_ACC);
    unsigned*           H1    = (unsigned*)(ws + OFF_H1);
    unsigned*           H2    = (unsigned*)(ws + OFF_H2);
    unsigned*           H3    = (unsigned*)(ws + OFF_H3);
    unsigned*           cnt   = (unsigned*)(ws + OFF_BINS);
    unsigned*           asum  = cnt + 16;
    unsigned long long* csum  = (unsigned long long*)(ws + OFF_BINS + 128);
    unsigned*           pfx   = (unsigned*)(ws + OFF_PFX);
    unsigned*           rr    = (unsigned*)(ws + OFF_RR);
    float*              edges = (float*)(ws + OFF_EDGES);

    ece_init <<<128,        256, 0, stream>>>(H1, pfx, rr);
    ece_rows <<<NROWS / 64, 256, 0, stream>>>(logits, labels, conf, acc);  // 15625 blocks
    ece_hist1<<<1024,       256, 0, stream>>>(conf, H1);
    ece_scan1<<<1,          256, 0, stream>>>(H1, pfx, rr);
    ece_hist2<<<1024,       256, 0, stream>>>(conf, pfx, H2);
    ece_scan2<<<NTGT,       256, 0, stream>>>(H2, pfx, rr);
    ece_hist3<<<1024,       256, 0, stream>>>(conf, pfx, H3);
    ece_scan3<<<NTGT,       256, 0, stream>>>(H3, pfx, rr);
    ece_edges<<<1,           32, 0, stream>>>(pfx, edges);
    ece_bin  <<<1024,       256, 0, stream>>>(conf, acc, edges, cnt, asum, csum);
    ece_final<<<1,            1, 0, stream>>>(cnt, asum, csum, (float*)d_out);
}